// SimpleQNN_18811956756655
// MI455X (gfx1250) — compile-verified
//
#include <hip/hip_runtime.h>
#include <hip/hip_bf16.h>

typedef __attribute__((ext_vector_type(2))) float v2f;
typedef __attribute__((ext_vector_type(8))) float v8f;

#define N_WIRES 16
#define BATCH   64
#define N_OUT   10

// out[b,o] = sum_w q[b,w] * W[o,w] + bias[o], where
// q[b,w] = prod_u ( maskbit ? z[b,u] : s[b,u] )
// z[b,u] = cos(rx_u) * (a^2 - b^2), s[b,u] = a^2 + b^2 with
// (a,b) = RY(ry_u) * H * RY(x_bu) |0>.
// Diagonal layers (CRZ, CZ, RZ) have unit-modulus entries and cancel in |psi|^2;
// the CNOT layer is a GF(2)-linear basis permutation folded into per-wire masks.
__global__ __launch_bounds__(256) void qnn_collapsed_kernel(
    const float* __restrict__ x,     // (64,16)
    const float* __restrict__ ry,    // (16)
    const float* __restrict__ rx,    // (16)
    const float* __restrict__ W,     // (10,16)
    const float* __restrict__ bias,  // (10)
    float* __restrict__ out)         // (64,10)
{
    __shared__ float    zArr[BATCH][N_WIRES];
    __shared__ float    sArr[BATCH][N_WIRES];
    __shared__ float    qArr[BATCH][N_WIRES];
    __shared__ unsigned wmask[N_WIRES];

    const int tid = threadIdx.x;

    // ---- Phase 0: per-wire masks from the CNOT-layer permutation h = g^{-1}.
    // h(e_k) = f15(...f1(f0(e_k))), f_w: flip bit(15-((w+1)&15)) if bit(15-w) set.
    if (tid < N_WIRES) {
        unsigned wm = 0u;
        for (int k = 0; k < 16; ++k) {
            unsigned c2 = (1u << k);
            for (int g = 0; g < 16; ++g) {
                const int cpos = 15 - g;
                const int tpos = 15 - ((g + 1) & 15);
                c2 ^= ((c2 >> cpos) & 1u) << tpos;
            }
            if ((c2 >> (15 - tid)) & 1u) wm |= (1u << (15 - k)); // bit pos k -> wire 15-k
        }
        wmask[tid] = wm;
    }

    // ---- Phase 1: per (batch, wire) Z-expectation z and norm s of the
    // single-qubit product-state factor after RX*RY*H*RY(x) |0>.
    const float RSQRT2 = 0.70710678118654752440f;
    for (int idx = tid; idx < BATCH * N_WIRES; idx += 256) {
        const int b = idx >> 4;
        const int w = idx & 15;
        const float xv  = x[idx];
        const float cxv = cosf(0.5f * xv);
        const float sxv = sinf(0.5f * xv);
        const float h0  = (cxv + sxv) * RSQRT2;   // H * RY(x)|0>
        const float h1  = (cxv - sxv) * RSQRT2;
        const float cy  = cosf(0.5f * ry[w]);
        const float sy  = sinf(0.5f * ry[w]);
        const float av  = cy * h0 - sy * h1;      // RY(ry) applied
        const float bv  = sy * h0 + cy * h1;
        const float a2  = av * av;
        const float b2  = bv * bv;
        // RX(rx): |top|^2 = c^2 a^2 + s^2 b^2, |bot|^2 = s^2 a^2 + c^2 b^2
        zArr[b][w] = cosf(rx[w]) * (a2 - b2);
        sArr[b][w] = a2 + b2;
    }
    __syncthreads();

    // ---- Phase 2: factorized expectation values q[b][w].
    for (int idx = tid; idx < BATCH * N_WIRES; idx += 256) {
        const int b = idx >> 4;
        const int w = idx & 15;
        const unsigned wm = wmask[w];
        float p = 1.0f;
#pragma unroll
        for (int u = 0; u < 16; ++u)
            p *= ((wm >> u) & 1u) ? zArr[b][u] : sArr[b][u];
        qArr[b][w] = p;
    }
    __syncthreads();

    // ---- Phase 3: out = q(64x16) @ W^T(16x10) + bias via V_WMMA_F32_16X16X4_F32.
    // Waves 0..3 each own one 16-row M-tile; chain 4 K=4 WMMA steps, bias in C.
    if (tid < 128) {
        const int wave = tid >> 5;        // M-tile index (uniform per wave)
        const int lane = tid & 31;
        const int half = lane >> 4;       // 0: lanes 0-15, 1: lanes 16-31
        const int col  = lane & 15;       // N index (B/C/D layout)
        const int grow = wave * 16 + (lane & 15); // A-matrix row (M) for this lane

        const float bb = (col < N_OUT) ? bias[col] : 0.0f;
        v8f c;
#pragma unroll
        for (int r = 0; r < 8; ++r) c[r] = bb;

#pragma unroll
        for (int kc = 0; kc < 4; ++kc) {
            // A 16x4 layout: lanes 0-15 -> K = k0,k0+1 ; lanes 16-31 -> K = k0+2,k0+3
            const int k0 = kc * 4 + 2 * half;
            v2f a, bm;
            a.x  = qArr[grow][k0];
            a.y  = qArr[grow][k0 + 1];
            // B 4x16 (K x N), B[k][n] = W[n][k]; pad columns >= 10 with zeros
            bm.x = (col < N_OUT) ? W[col * N_WIRES + k0]     : 0.0f;
            bm.y = (col < N_OUT) ? W[col * N_WIRES + k0 + 1] : 0.0f;
            c = __builtin_amdgcn_wmma_f32_16x16x4_f32(
                    /*neg_a=*/false, a, /*neg_b=*/false, bm,
                    /*c_mod=*/(short)0, c, /*reuse_a=*/false, /*reuse_b=*/false);
        }

        // D layout: VGPR r, lanes 0-15 -> M=r, lanes 16-31 -> M=r+8; N = col
#pragma unroll
        for (int r = 0; r < 8; ++r) {
            const int m = wave * 16 + r + 8 * half;
            if (col < N_OUT) out[m * N_OUT + col] = c[r];
        }
    }
}

extern "C" void kernel_launch(void* const* d_in, const int* in_sizes, int n_in,
                              void* d_out, int out_size, void* d_ws, size_t ws_size,
                              hipStream_t stream) {
    const float* x  = (const float*)d_in[0]; // (64,16)
    const float* ry = (const float*)d_in[1]; // (16)
    const float* rx = (const float*)d_in[2]; // (16)
    // d_in[3] = rz_params, d_in[4] = crz_params: unit-modulus diagonal phases,
    // provably cancel in |psi|^2 -> unused.
    const float* W  = (const float*)d_in[5]; // (10,16)
    const float* b  = (const float*)d_in[6]; // (10)
    qnn_collapsed_kernel<<<1, 256, 0, stream>>>(x, ry, rx, W, b, (float*)d_out);
}